// QuantumSSM_27917287424286
// MI455X (gfx1250) — compile-verified
//
#include <hip/hip_runtime.h>
#include <math.h>

typedef __attribute__((ext_vector_type(2)))  float v2f;
typedef __attribute__((ext_vector_type(8)))  float v8f;

#define NQ    4
#define TLEN  512
#define BATCH 256
#define IDIM  256
#define HD    256

// ---------------------------------------------------------------------------
// Kernel A: yx[m][q] = entry_b[q] + sum_i x[m][i]*exp(fw[i])*entry_W[q][i]
// M = B*T = 131072 rows, K = 256, N padded 4 -> 16.  WMMA f32 16x16x4:
//   A(16x4)  = scaled Wx chunk from a zero-padded 16-row LDS tile (no EXEC
//              divergence): lanes 0-15: M=lane, K=0/1 in v0/v1; lanes 16-31:
//              same M, K=2/3.
//   B(4x16)  = x chunk (non-temporal: streamed once), mirrored K layout,
//              N = sample row.
//   D(16x16) : lanes 0-15 hold D[q][lane] in c[q].
// One wave per 16-row tile, 8 waves per block -> 1024 blocks.
// ---------------------------------------------------------------------------
__global__ void __launch_bounds__(256) entry_precompute(
    const float* __restrict__ x, const float* __restrict__ fw,
    const float* __restrict__ entryW, const float* __restrict__ entryB,
    float* __restrict__ yx) {
  __shared__ float Wxs[16 * IDIM];         // rows 0-3: exp(fw)*Wx, rows 4-15: 0
  const int tid = threadIdx.x;
  {
    float e = __expf(fw[tid]);
#pragma unroll
    for (int q = 0; q < 16; ++q)
      Wxs[q * IDIM + tid] = (q < 4) ? entryW[q * 512 + tid] * e : 0.f;
  }
  __syncthreads();

  const int wave = tid >> 5;
  const int lane = tid & 31;
  const int m    = lane & 15;                          // A row / B col in tile
  const int kofs = (lane >> 4) * 2;                    // K sub-offset per half-wave
  const long row_base = ((long)blockIdx.x * 8 + wave) * 16;
  const float* xrow  = x + (row_base + m) * IDIM + kofs;
  const float* wsrow = Wxs + m * IDIM + kofs;

  v8f c = {};
#pragma unroll 8
  for (int kk = 0; kk < IDIM; kk += 4) {
    v2f a = *(const v2f*)(wsrow + kk);                           // LDS, padded
    v2f b = __builtin_nontemporal_load((const v2f*)(xrow + kk)); // stream x
    c = __builtin_amdgcn_wmma_f32_16x16x4_f32(false, a, false, b, (short)0, c,
                                              false, false);
  }

  if (lane < 16) {                                     // rows q=0..3 live in c[0..3]
    float4 r;
    r.x = c[0] + entryB[0];
    r.y = c[1] + entryB[1];
    r.z = c[2] + entryB[2];
    r.w = c[3] + entryB[3];
    *(float4*)(yx + (row_base + lane) * 4) = r;        // re-read by scan: keep RT
  }
}

// ---------------------------------------------------------------------------
// 4-qubit state-vector gates: one amplitude per lane (idx = lane & 15),
// partner exchange via shfl_xor.  Qubit i toggles bit (8 >> i).
// ---------------------------------------------------------------------------
__device__ __forceinline__ void gate_h(float& ar, float& ai, int idx, int mask) {
  const float R = 0.70710678118654752f;
  float pr = __shfl_xor(ar, mask, 32);
  float pi = __shfl_xor(ai, mask, 32);
  float cs = (idx & mask) ? -R : R;
  ar = cs * ar + R * pr;
  ai = cs * ai + R * pi;
}
__device__ __forceinline__ void gate_ry(float& ar, float& ai, int idx, int mask,
                                        float c, float s) {
  float pr = __shfl_xor(ar, mask, 32);
  float pi = __shfl_xor(ai, mask, 32);
  float cp = (idx & mask) ? s : -s;
  float nr = c * ar + cp * pr;
  float ni = c * ai + cp * pi;
  ar = nr; ai = ni;
}
__device__ __forceinline__ void gate_rz(float& ar, float& ai, int idx, int mask,
                                        float c, float s) {
  float ss = (idx & mask) ? s : -s;    // bit0: e^{-i t/2}, bit1: e^{+i t/2}
  float nr = c * ar - ss * ai;
  float ni = c * ai + ss * ar;
  ar = nr; ai = ni;
}
__device__ __forceinline__ void gate_rx(float& ar, float& ai, int idx, int mask,
                                        float c, float s) {
  float pr = __shfl_xor(ar, mask, 32);
  float pi = __shfl_xor(ai, mask, 32);
  float nr = c * ar + s * pi;          // -i*s applied to partner
  float ni = c * ai - s * pr;
  ar = nr; ai = ni;
}
__device__ __forceinline__ void gate_cnot(float& ar, float& ai, int idx,
                                          int cmask, int tmask) {
  float pr = __shfl_xor(ar, tmask, 32);
  float pi = __shfl_xor(ai, tmask, 32);
  if (idx & cmask) { ar = pr; ai = pi; }
}
__device__ __forceinline__ float wave_reduce(float v) {
  v += __shfl_xor(v, 1, 32);
  v += __shfl_xor(v, 2, 32);
  v += __shfl_xor(v, 4, 32);
  v += __shfl_xor(v, 8, 32);
  v += __shfl_xor(v, 16, 32);
  return v;
}

// ---------------------------------------------------------------------------
// Kernel B: the sequential scan.  One block (8 wave32) per batch sample.
// Per step: 2 barriers.  Wave 0 runs the quantum circuit on its 32 lanes.
// ---------------------------------------------------------------------------
__global__ void __launch_bounds__(256) qssm_scan(
    const float* __restrict__ yx, const float* __restrict__ entryW,
    const float* __restrict__ vqcW, const float* __restrict__ exitW,
    const float* __restrict__ exitB, float* __restrict__ outs,
    float* __restrict__ hT) {
  const int b    = blockIdx.x;
  const int tid  = threadIdx.x;
  const int wave = tid >> 5;
  const int lane = tid & 31;

  __shared__ float4 part[8];     // per-wave partial sums of h . Wh^T
  __shared__ float  sbuf[4];     // y + <Z>
  __shared__ float  cw[24], sw[24];

  if (tid < 24) {                // fixed variational half-angle trig
    float w = 0.5f * vqcW[tid];  // layout [2][3][4] -> l*12 + rot*4 + qubit
    cw[tid] = __cosf(w);
    sw[tid] = __sinf(w);
  }

  const float wh0 = entryW[0 * 512 + 256 + tid];
  const float wh1 = entryW[1 * 512 + 256 + tid];
  const float wh2 = entryW[2 * 512 + 256 + tid];
  const float wh3 = entryW[3 * 512 + 256 + tid];
  const float4 ewg = *(const float4*)(exitW + (long)tid * 4);
  const float4 ewd = *(const float4*)(exitW + (long)(tid + 256) * 4);
  const float  ebg = exitB[tid];
  const float  ebd = exitB[tid + 256];

  float h = 0.f;
  const float4* yx4  = (const float4*)yx + (long)b * TLEN;
  float*        outp = outs + (long)b * TLEN * HD + tid;

  __syncthreads();

  for (int t = 0; t < TLEN; ++t) {
    // ---- h @ Wh^T : per-wave reduction, cross-wave combine in LDS ----
    float p0 = wave_reduce(h * wh0);
    float p1 = wave_reduce(h * wh1);
    float p2 = wave_reduce(h * wh2);
    float p3 = wave_reduce(h * wh3);
    if (lane == 0) part[wave] = make_float4(p0, p1, p2, p3);
    __syncthreads();

    if (wave == 0) {
      float4 acc = yx4[t];                 // precomputed x-projection (+bias)
#pragma unroll
      for (int w2 = 0; w2 < 8; ++w2) {
        float4 p = part[w2];
        acc.x += p.x; acc.y += p.y; acc.z += p.z; acc.w += p.w;
      }
      float y[4] = {acc.x, acc.y, acc.z, acc.w};

      // ---- 4-qubit circuit: amplitude idx per lane ----
      int   idx = lane & 15;
      float ar  = (idx == 0) ? 1.f : 0.f, ai = 0.f;
#pragma unroll
      for (int i = 0; i < 4; ++i) {        // encoding: H, RY(atan y), RZ(atan y^2)
        int mask = 8 >> i;
        gate_h(ar, ai, idx, mask);
        float ta = 0.5f * atanf(y[i]);
        gate_ry(ar, ai, idx, mask, __cosf(ta), __sinf(ta));
        float tz = 0.5f * atanf(y[i] * y[i]);
        gate_rz(ar, ai, idx, mask, __cosf(tz), __sinf(tz));
      }
#pragma unroll
      for (int l = 0; l < 2; ++l) {        // ansatz layers
#pragma unroll
        for (int i = 0; i < 4; ++i)
          gate_cnot(ar, ai, idx, 8 >> i, 8 >> ((i + 1) & 3));
#pragma unroll
        for (int i = 0; i < 4; ++i) {
          gate_rx(ar, ai, idx, 8 >> i, cw[l * 12 + 0 + i], sw[l * 12 + 0 + i]);
          gate_ry(ar, ai, idx, 8 >> i, cw[l * 12 + 4 + i], sw[l * 12 + 4 + i]);
          gate_rz(ar, ai, idx, 8 >> i, cw[l * 12 + 8 + i], sw[l * 12 + 8 + i]);
        }
      }
      // ---- <Z_i> = sum_amp sign_i * |amp|^2 ----
      float prob = ar * ar + ai * ai;
      float z[4];
#pragma unroll
      for (int i = 0; i < 4; ++i) {
        float v = (idx & (8 >> i)) ? -prob : prob;
        v += __shfl_xor(v, 1, 32);
        v += __shfl_xor(v, 2, 32);
        v += __shfl_xor(v, 4, 32);
        v += __shfl_xor(v, 8, 32);
        z[i] = v;
      }
      if (lane == 0) {
        sbuf[0] = y[0] + z[0];
        sbuf[1] = y[1] + z[1];
        sbuf[2] = y[2] + z[2];
        sbuf[3] = y[3] + z[3];
      }
    }
    __syncthreads();

    // ---- exit projection + gated update, one output element per thread ----
    float s0 = sbuf[0], s1 = sbuf[1], s2 = sbuf[2], s3 = sbuf[3];
    float og = s0 * ewg.x + s1 * ewg.y + s2 * ewg.z + s3 * ewg.w + ebg;
    float od = s0 * ewd.x + s1 * ewd.y + s2 * ewd.z + s3 * ewd.w + ebd;
    float g  = 1.f / (1.f + __expf(-og));
    g = fminf(fmaxf(g, 0.05f), 0.95f);
    h = (1.f - g) * h + g * tanhf(od);
    __builtin_nontemporal_store(h, outp + (long)t * HD);  // written once, never re-read
  }
  hT[(long)b * HD + tid] = h;
}

// ---------------------------------------------------------------------------
extern "C" void kernel_launch(void* const* d_in, const int* in_sizes, int n_in,
                              void* d_out, int out_size, void* d_ws, size_t ws_size,
                              hipStream_t stream) {
  const float* x      = (const float*)d_in[0];   // [B,T,I]
  const float* fw     = (const float*)d_in[1];   // [1,1,I]
  const float* entryW = (const float*)d_in[2];   // [4, I+H]
  const float* entryB = (const float*)d_in[3];   // [4]
  const float* vqcW   = (const float*)d_in[4];   // [2,3,4]
  const float* exitW  = (const float*)d_in[5];   // [2H, 4]
  const float* exitB  = (const float*)d_in[6];   // [2H]

  float* out  = (float*)d_out;
  float* outs = out;                                   // [B,T,H]
  float* hT   = out + (long)BATCH * TLEN * HD;         // [B,H]
  float* yx   = (float*)d_ws;                          // [B*T,4] scratch (2 MB)

  // 131072 rows / (8 waves * 16 rows) = 1024 blocks
  entry_precompute<<<(BATCH * TLEN) / 128, 256, 0, stream>>>(x, fw, entryW,
                                                             entryB, yx);
  qssm_scan<<<BATCH, 256, 0, stream>>>(yx, entryW, vqcW, exitW, exitB, outs, hT);
}